// MatrixCapsEM_74612171866726
// MI455X (gfx1250) — compile-verified
//
#include <hip/hip_runtime.h>
#include <math.h>

typedef float v2f __attribute__((ext_vector_type(2)));
typedef float v8f __attribute__((ext_vector_type(8)));

#define I_CAPS 1024
#define J_CAPS 64
#define HSZ    16
#define BATCH  64

// ---------------------------------------------------------------------------
// Kernel 1: vijh[b,i,j,h] = sum_y pi[b,i,x,y] * w[i,j,y,n]   (h = x*4+n)
// Per i: a 256x4x256 fp32 GEMM -> 16x16 tiles of V_WMMA_F32_16X16X4_F32.
// All 256 tiles (256x256 f32 = 256KB) are staged in the 320KB WGP LDS, then
// written back as fully coalesced b128 stores (4KB contiguous per batch row).
// ---------------------------------------------------------------------------
__global__ void __launch_bounds__(256) caps_vij_wmma(
    const float* __restrict__ xi,      // (B, 17, 1024)
    const float* __restrict__ weight,  // (1024, 64, 4, 4)
    float* __restrict__ v)             // (B, 1024, 64, 16)
{
  __shared__ float Dsh[256 * 256];     // 256 KB: full D matrix for this i

  const int i    = blockIdx.x;
  const int lane = threadIdx.x & 31;
  const int wave = threadIdx.x >> 5;
  const int half = lane >> 4;   // K-half select (A/B frags) / M-half (D frag)
  const int m16  = lane & 15;   // row within row-tile (A) / col within col-tile (B,D)

  for (int e = 0; e < 2; ++e) {
    const int tm = wave * 2 + e;      // row tile 0..15
    const int r0 = tm * 16;
    const int r  = r0 + m16;          // global row = b*4 + x
    const int b  = r >> 2;
    const int x  = r & 3;
    // A fragment: A[m][k], k = 2*half + q  (pi[b,i,x,y] = xi[b, x*4+y, i])
    v2f afrag;
    afrag.x = xi[((size_t)b * 17 + (x * 4 + 2 * half + 0)) * I_CAPS + i];
    afrag.y = xi[((size_t)b * 17 + (x * 4 + 2 * half + 1)) * I_CAPS + i];

    for (int tn = 0; tn < 16; ++tn) {
      const int c0 = tn * 16;
      const int c  = c0 + m16;        // global col = j*4 + n
      const int j  = c >> 2;
      const int n  = c & 3;
      // B fragment: B[k][n_col], k = 2*half + q
      v2f bfrag;
      bfrag.x = weight[(((size_t)i * J_CAPS + j) * 4 + (2 * half + 0)) * 4 + n];
      bfrag.y = weight[(((size_t)i * J_CAPS + j) * 4 + (2 * half + 1)) * 4 + n];

      v8f acc = {};
      acc = __builtin_amdgcn_wmma_f32_16x16x4_f32(
          /*neg_a=*/false, afrag, /*neg_b=*/false, bfrag,
          /*c_mod=*/(short)0, acc, /*reuse_a=*/false, /*reuse_b=*/false);

      // D layout: VGPR q holds row (q + 8*half), col = m16 within tile
      #pragma unroll
      for (int q = 0; q < 8; ++q) {
        const int gr = r0 + q + 8 * half;
        Dsh[gr * 256 + c] = acc[q];
      }
    }
  }
  __syncthreads();

  // Coalesced writeback: v element (bb, i, jj, h=xx*4+nn) = Dsh[bb*4+xx][jj*4+nn]
  // One k-iteration = one bb = 256 consecutive float4 = 4KB contiguous.
  const int jj = threadIdx.x >> 2;   // 0..63
  const int xx = threadIdx.x & 3;    // 0..3
  for (int bb = 0; bb < BATCH; ++bb) {
    const float4 val = *(const float4*)&Dsh[(bb * 4 + xx) * 256 + jj * 4];
    *(float4*)&v[(((size_t)bb * I_CAPS + i) * J_CAPS + jj) * HSZ + xx * 4] = val;
  }
}

// ---------------------------------------------------------------------------
// Kernel 2: EM routing. One 1024-thread workgroup per batch element.
// Full R (1024x64 fp32 = 256KB) resident in CDNA5's 320KB WGP LDS.
// ---------------------------------------------------------------------------
__global__ void __launch_bounds__(1024) caps_em_routing(
    const float* __restrict__ xi,   // (B, 17, 1024)
    const float* __restrict__ v,    // (B, 1024, 64, 16)
    const float* __restrict__ beta_a_p,
    const float* __restrict__ beta_u_p,
    float* __restrict__ out)        // (B, 17, 64)
{
  __shared__ float Rsh[I_CAPS * J_CAPS];   // 256 KB
  __shared__ float ai_sh[I_CAPS];          // 4 KB
  __shared__ float mu_sh[J_CAPS * HSZ];    // 4 KB
  __shared__ float var_sh[J_CAPS * HSZ];   // 4 KB
  __shared__ float i2v_sh[J_CAPS * HSZ];   // 4 KB : 1/(2*var)
  __shared__ float rowsum_sh[I_CAPS];      // 4 KB
  __shared__ float sumR_sh[J_CAPS];
  __shared__ float aj_sh[J_CAPS];
  __shared__ float lpc_sh[J_CAPS];

  const int b   = blockIdx.x;
  const int tid = threadIdx.x;
  const float beta_a = beta_a_p[0];
  const float beta_u = beta_u_p[0];
  const float* __restrict__ vb = v + (size_t)b * I_CAPS * J_CAPS * HSZ;

  ai_sh[tid] = xi[((size_t)b * 17 + 16) * I_CAPS + tid];
  for (int k = 0; k < 64; ++k) Rsh[tid + k * 1024] = 1.0f / (float)J_CAPS;
  __syncthreads();

  const int j = tid >> 4;    // 0..63
  const int h = tid & 15;    // 0..15

  for (int it = 0; it < 3; ++it) {
    // ---- R *= a_i ----
    for (int k = 0; k < 64; ++k) {
      const int idx = tid + k * 1024;
      Rsh[idx] *= ai_sh[idx >> 6];
    }
    __syncthreads();

    // ---- sumR[j], mu[j][h]  (coalesced: 32 lanes cover 128B of v) ----
    float smu = 0.f, sR = 0.f;
    for (int i = 0; i < I_CAPS; ++i) {
      const float Rv = Rsh[i * J_CAPS + j];
      smu += Rv * vb[((size_t)i * J_CAPS + j) * HSZ + h];
      sR  += Rv;
    }
    mu_sh[tid] = smu;
    if (h == 0) sumR_sh[j] = sR;
    __syncthreads();
    const float sr = sumR_sh[j];
    const float mu = mu_sh[tid] / sr;
    mu_sh[tid] = mu;
    __syncthreads();

    // ---- var[j][h] ----
    float sv = 0.f;
    for (int i = 0; i < I_CAPS; ++i) {
      const float d = vb[((size_t)i * J_CAPS + j) * HSZ + h] - mu;
      sv += Rsh[i * J_CAPS + j] * d * d;
    }
    const float vv = sv / sr + 1e-4f;
    var_sh[tid] = vv;
    i2v_sh[tid] = 0.5f / vv;
    __syncthreads();

    // ---- a_j, log-normalizer ----
    if (tid < J_CAPS) {
      float csum = 0.f, lsum = 0.f;
      const float srj = sumR_sh[tid];
      for (int hh = 0; hh < HSZ; ++hh) {
        const float vq = var_sh[tid * HSZ + hh];
        csum += (beta_u + 0.5f * logf(vq)) * srj;
        lsum += logf(6.28318530717958647692f * vq);
      }
      aj_sh[tid]  = 1.0f / (1.0f + expf(-(beta_a - csum)));
      lpc_sh[tid] = 0.5f * lsum;   // log sqrt(prod 2*pi*var)
    }
    __syncthreads();

    // ---- new R (unnormalized): a_j * exp(log_p - log_norm), b128 loads ----
    for (int k = 0; k < 64; ++k) {
      const int idx = tid + k * 1024;
      const int ii = idx >> 6;
      const int jj = idx & 63;
      const float4* __restrict__ vp4 =
          (const float4*)(vb + ((size_t)ii * J_CAPS + jj) * HSZ);
      float vl[16];
      *(float4*)&vl[0]  = vp4[0];
      *(float4*)&vl[4]  = vp4[1];
      *(float4*)&vl[8]  = vp4[2];
      *(float4*)&vl[12] = vp4[3];
      float lp = 0.f;
      #pragma unroll
      for (int hh = 0; hh < HSZ; ++hh) {
        const float d = vl[hh] - mu_sh[jj * HSZ + hh];
        lp += d * d * i2v_sh[jj * HSZ + hh];
      }
      Rsh[idx] = aj_sh[jj] * expf(-lp - lpc_sh[jj]);
    }
    __syncthreads();

    // ---- softmax normalization over j (entirely in LDS) ----
    {
      float rs = 0.f;
      for (int jj = 0; jj < J_CAPS; ++jj) rs += Rsh[tid * J_CAPS + jj];
      rowsum_sh[tid] = rs;
    }
    __syncthreads();
    for (int k = 0; k < 64; ++k) {
      const int idx = tid + k * 1024;
      Rsh[idx] /= rowsum_sh[idx >> 6];
    }
    __syncthreads();
  }

  // ---- pj[j][h] = sum_i v*R ; output (B,17,J) ----
  float s = 0.f;
  for (int i = 0; i < I_CAPS; ++i)
    s += vb[((size_t)i * J_CAPS + j) * HSZ + h] * Rsh[i * J_CAPS + j];
  out[((size_t)b * 17 + h) * J_CAPS + j] = s;
  if (tid < J_CAPS)
    out[((size_t)b * 17 + 16) * J_CAPS + tid] = aj_sh[tid];
}

// ---------------------------------------------------------------------------
extern "C" void kernel_launch(void* const* d_in, const int* in_sizes, int n_in,
                              void* d_out, int out_size, void* d_ws, size_t ws_size,
                              hipStream_t stream) {
  const float* xi     = (const float*)d_in[0];  // 64*17*1024
  const float* weight = (const float*)d_in[1];  // 1024*64*4*4
  const float* beta_a = (const float*)d_in[2];
  const float* beta_u = (const float*)d_in[3];
  float* out = (float*)d_out;                   // 64*17*64
  float* v   = (float*)d_ws;                    // needs 64*1024*64*16*4 = 256 MB

  caps_vij_wmma<<<dim3(I_CAPS), 256, 0, stream>>>(xi, weight, v);
  caps_em_routing<<<dim3(BATCH), 1024, 0, stream>>>(xi, v, beta_a, beta_u, out);
}